// GNNModel_20272245637468
// MI455X (gfx1250) — compile-verified
//
#include <hip/hip_runtime.h>
#include <hip/hip_bf16.h>

// ---------------------------------------------------------------------------
// 2-layer GCN for MI455X (gfx1250).
//  - Dense GEMMs: V_WMMA_F32_16X16X4_F32, fp32 in/out (matches reference
//    dtype). One wave computes a full 16-row stripe across all N-tiles so
//    each A fragment feeds 6 (layer1) / 4 (layer2) WMMAs.
//  - Propagation: memory-bound gather + float atomicAdd scatter; working set
//    (2x19MB node buffers + 12.8MB edges) is L2-resident on the 192MB L2.
//    Roofline: ~1.3GB traffic -> ~60us at 23.3TB/s; GEMMs (~1.5 GFLOP) are
//    negligible on the WMMA pipe.
// ---------------------------------------------------------------------------

typedef __attribute__((ext_vector_type(2))) float vf2;
typedef __attribute__((ext_vector_type(8))) float vf8;

#define N_NODES 50000
#define D_IN    96
#define D_HID   96
#define D_OUT   64

// ---------------- degree / normalization ----------------

__global__ void deg_init_kernel(float* __restrict__ deg, int n) {
    int i = blockIdx.x * blockDim.x + threadIdx.x;
    if (i < n) deg[i] = 1.0f;  // self loop
}

__global__ void deg_edge_kernel(const int* __restrict__ dst,
                                float* __restrict__ deg, int e_cnt) {
    int e = blockIdx.x * blockDim.x + threadIdx.x;
    if (e < e_cnt) atomicAdd(&deg[dst[e]], 1.0f);
}

__global__ void dinv_kernel(float* __restrict__ deg_dinv, int n) {
    int i = blockIdx.x * blockDim.x + threadIdx.x;
    if (i < n) deg_dinv[i] = rsqrtf(deg_dinv[i]);  // deg >= 1 always
}

__global__ void norm_kernel(const int* __restrict__ src,
                            const int* __restrict__ dst,
                            const float* __restrict__ dinv,
                            float* __restrict__ norm, int e_cnt) {
    int e = blockIdx.x * blockDim.x + threadIdx.x;
    if (e < e_cnt) norm[e] = dinv[src[e]] * dinv[dst[e]];
}

// ---------------- WMMA fp32 GEMM: C[M,16*NT] = A[M,K] @ W[K,16*NT] --------
// One wave per 16-row stripe; computes all NT 16x16 tiles of that stripe.
// A 16x4 layout: lanes 0-15 -> M=lane, K={k,k+1}; lanes 16-31 -> K={k+2,k+3}.
// B 4x16 mirrored over columns. C/D: VGPR r -> row r (lanes 0-15), r+8 (hi).

template <int K, int NT>
__global__ __launch_bounds__(32)
void gemm_wmma_f32_kernel(const float* __restrict__ A,
                          const float* __restrict__ W,
                          float* __restrict__ C) {
    constexpr int N = NT * 16;
    const int lane = threadIdx.x;
    const int half = lane >> 4;       // 0: lanes 0-15, 1: lanes 16-31
    const int l    = lane & 15;
    const int tm   = blockIdx.x;      // M tile
    const float* __restrict__ arow = A + (size_t)(tm * 16 + l) * K;

    vf8 acc[NT] = {};
#pragma unroll
    for (int k = 0; k < K; k += 4) {
        const int ka = k + half * 2;
        vf2 a = *reinterpret_cast<const vf2*>(arow + ka);
#pragma unroll
        for (int t = 0; t < NT; ++t) {
            const int col = t * 16 + l;
            vf2 b;
            b[0] = W[(ka + 0) * N + col];
            b[1] = W[(ka + 1) * N + col];
            acc[t] = __builtin_amdgcn_wmma_f32_16x16x4_f32(
                /*neg_a=*/false, a, /*neg_b=*/false, b,
                /*c_mod=*/(short)0, acc[t], /*reuse_a=*/false, /*reuse_b=*/false);
        }
    }
#pragma unroll
    for (int t = 0; t < NT; ++t) {
        const int col = t * 16 + l;
#pragma unroll
        for (int r = 0; r < 8; ++r) {
            C[(size_t)(tm * 16 + r + half * 8) * N + col] = acc[t][r];
        }
    }
}

// ---------------- propagation ----------------

// out[i,c] = dinv[i]^2 * H[i,c] + bias[c]   (self-loop term + bias folded in)
template <int COLS>
__global__ void self_bias_kernel(const float* __restrict__ H,
                                 const float* __restrict__ dinv,
                                 const float* __restrict__ bias,
                                 float* __restrict__ out,
                                 int total4) {
    constexpr int C4 = COLS / 4;
    int idx = blockIdx.x * blockDim.x + threadIdx.x;
    if (idx >= total4) return;
    int i = idx / C4;
    int c = (idx % C4) * 4;
    float di = dinv[i];
    float s  = di * di;
    const float4 h = *reinterpret_cast<const float4*>(H + (size_t)i * COLS + c);
    const float4 b = *reinterpret_cast<const float4*>(bias + c);
    float4 o;
    o.x = s * h.x + b.x;
    o.y = s * h.y + b.y;
    o.z = s * h.z + b.z;
    o.w = s * h.w + b.w;
    *reinterpret_cast<float4*>(out + (size_t)i * COLS + c) = o;
}

// out[dst, c..c+3] += norm[e] * H[src, c..c+3]
template <int COLS>
__global__ void scatter_kernel(const float* __restrict__ H,
                               const int* __restrict__ src,
                               const int* __restrict__ dst,
                               const float* __restrict__ norm,
                               float* __restrict__ out,
                               int total4) {
    constexpr int C4 = COLS / 4;
    int idx = blockIdx.x * blockDim.x + threadIdx.x;
    if (idx >= total4) return;
    int e = idx / C4;
    int c = (idx % C4) * 4;
    int s = src[e];
    int d = dst[e];
    float w = norm[e];
    const float4 h = *reinterpret_cast<const float4*>(H + (size_t)s * COLS + c);
    float* o = out + (size_t)d * COLS + c;
    atomicAdd(o + 0, w * h.x);
    atomicAdd(o + 1, w * h.y);
    atomicAdd(o + 2, w * h.z);
    atomicAdd(o + 3, w * h.w);
}

__global__ void relu_kernel(const float* __restrict__ in,
                            float* __restrict__ out, int total) {
    int idx = blockIdx.x * blockDim.x + threadIdx.x;
    if (idx < total) out[idx] = fmaxf(in[idx], 0.0f);
}

// ---------------- launcher ----------------

static inline int cdiv(long long a, int b) { return (int)((a + b - 1) / b); }

extern "C" void kernel_launch(void* const* d_in, const int* in_sizes, int n_in,
                              void* d_out, int out_size, void* d_ws, size_t ws_size,
                              hipStream_t stream) {
    const float* x   = (const float*)d_in[0];
    const float* W1  = (const float*)d_in[1];
    const float* b1  = (const float*)d_in[2];
    const float* W2  = (const float*)d_in[3];
    const float* b2  = (const float*)d_in[4];
    const int*   ei  = (const int*)d_in[5];   // (2, E) int32

    const int n = N_NODES;
    const int e_cnt = in_sizes[5] / 2;        // 800000
    const int* src = ei;
    const int* dst = ei + e_cnt;
    float* out = (float*)d_out;

    // Workspace layout (floats). All buffer starts 16-byte aligned:
    // dinv at 0, norm at n (50000*4 B), bufA at (n+e_cnt)=850000 floats
    // (3,400,000 B, /16 ok), bufB at +4.8M floats (22,600,000 B, /16 ok).
    float* ws   = (float*)d_ws;
    float* dinv = ws;                              // n
    float* norm = dinv + n;                        // e_cnt
    float* bufA = norm + e_cnt;                    // n*96 : H1, then relu(h)
    float* bufB = bufA + (size_t)n * D_HID;        // n*96 : agg1, then H2

    const int B = 256;

    // 1) degrees -> dinv -> per-edge norms
    deg_init_kernel<<<cdiv(n, B), B, 0, stream>>>(dinv, n);
    deg_edge_kernel<<<cdiv(e_cnt, B), B, 0, stream>>>(dst, dinv, e_cnt);
    dinv_kernel<<<cdiv(n, B), B, 0, stream>>>(dinv, n);
    norm_kernel<<<cdiv(e_cnt, B), B, 0, stream>>>(src, dst, dinv, norm, e_cnt);

    // 2) H1 = x @ W1  (50000x96 @ 96x96), WMMA fp32, 6 N-tiles per wave
    gemm_wmma_f32_kernel<D_IN, D_HID / 16>
        <<<n / 16, 32, 0, stream>>>(x, W1, bufA);

    // 3) layer-1 aggregation: self+bias into bufB, then edge scatter
    const int tot1_4 = n * (D_HID / 4);
    self_bias_kernel<D_HID><<<cdiv(tot1_4, B), B, 0, stream>>>(bufA, dinv, b1, bufB, tot1_4);
    const int sc1_4 = e_cnt * (D_HID / 4);
    scatter_kernel<D_HID><<<cdiv(sc1_4, B), B, 0, stream>>>(bufA, src, dst, norm, bufB, sc1_4);

    // 4) ReLU -> bufA now holds relu(h)
    const int tot1 = n * D_HID;
    relu_kernel<<<cdiv(tot1, B), B, 0, stream>>>(bufB, bufA, tot1);

    // 5) H2 = relu(h) @ W2  (50000x96 @ 96x64), WMMA fp32, 4 N-tiles per wave
    gemm_wmma_f32_kernel<D_HID, D_OUT / 16>
        <<<n / 16, 32, 0, stream>>>(bufA, W2, bufB);

    // 6) layer-2 aggregation directly into d_out (bias folded into init,
    //    every output element written each call)
    const int tot2_4 = n * (D_OUT / 4);
    self_bias_kernel<D_OUT><<<cdiv(tot2_4, B), B, 0, stream>>>(bufB, dinv, b2, out, tot2_4);
    const int sc2_4 = e_cnt * (D_OUT / 4);
    scatter_kernel<D_OUT><<<cdiv(sc2_4, B), B, 0, stream>>>(bufB, src, dst, norm, out, sc2_4);
}